// LocMotionBranch_65781719106365
// MI455X (gfx1250) — compile-verified
//
#include <hip/hip_runtime.h>
#include <hip/hip_bf16.h>
#include <math.h>

typedef __attribute__((ext_vector_type(2))) float v2f;
typedef __attribute__((ext_vector_type(8))) float v8f;

// ---------------------------------------------------------------------------
// 0) zero scratch accumulators
// ---------------------------------------------------------------------------
__global__ void k_zero(float* __restrict__ p, long n) {
    long i = (long)blockIdx.x * blockDim.x + threadIdx.x;
    if (i < n) p[i] = 0.0f;
}

// ---------------------------------------------------------------------------
// 1) superpixel pooling with LDS privatization.
//    grid = (chunks, B), block = 256, dyn-LDS = K*5 floats.
//    Per-pixel ds_add_f32 into acc[label][5], one global-atomic flush / block.
// ---------------------------------------------------------------------------
__global__ void k_pool(const int* __restrict__ labels,
                       const float* __restrict__ fx,
                       const float* __restrict__ fy,
                       float* __restrict__ sums,     // [B*K, 4]
                       float* __restrict__ counts,   // [B*K]
                       int K, int P, int side, int chunks) {
    extern __shared__ float acc[];  // K*5
    const int b = blockIdx.y;
    for (int i = threadIdx.x; i < K * 5; i += blockDim.x) acc[i] = 0.0f;
    __syncthreads();

    const int per = P / chunks;
    const int p0  = blockIdx.x * per;
    const int*   lab = labels + (long)b * P;
    const float* fxb = fx + (long)b * P;
    const float* fyb = fy + (long)b * P;
    const float invs = 2.0f / (float)(side - 1);

    for (int p = p0 + threadIdx.x; p < p0 + per; p += blockDim.x) {
        const int i2 = p / side, i3 = p - i2 * side;
        const float xx = (float)i2 * invs - 1.0f;
        const float yy = (float)i3 * invs - 1.0f;
        float* a = acc + lab[p] * 5;
        atomicAdd(a + 0, fxb[p]);
        atomicAdd(a + 1, fyb[p]);
        atomicAdd(a + 2, xx);
        atomicAdd(a + 3, yy);
        atomicAdd(a + 4, 1.0f);
    }
    __syncthreads();

    for (int s = threadIdx.x; s < K; s += blockDim.x) {
        const float c = acc[s * 5 + 4];
        if (c != 0.0f) {
            const long g = (long)b * K + s;
            atomicAdd(&sums[g * 4 + 0], acc[s * 5 + 0]);
            atomicAdd(&sums[g * 4 + 1], acc[s * 5 + 1]);
            atomicAdd(&sums[g * 4 + 2], acc[s * 5 + 2]);
            atomicAdd(&sums[g * 4 + 3], acc[s * 5 + 3]);
            atomicAdd(&counts[g], c);
        }
    }
}

// ---------------------------------------------------------------------------
// 2) build X[Ntok, 8] = [t, mean_fx, mean_fy, mean_xx, mean_yy, 0, 0, 0]
// ---------------------------------------------------------------------------
__global__ void k_buildX(const float* __restrict__ sums,
                         const float* __restrict__ counts,
                         const int* __restrict__ frame_idx,
                         const int* __restrict__ n_frames,
                         float* __restrict__ X, int K, int Ntok) {
    const int g = blockIdx.x * blockDim.x + threadIdx.x;
    if (g >= Ntok) return;
    const int b = g / K;
    const float t   = (float)frame_idx[b] / ((float)(*n_frames) - 1.0f);
    const float inv = 1.0f / fmaxf(counts[g], 1.0f);
    float* x = X + (long)g * 8;
    x[0] = t;
    x[1] = sums[(long)g * 4 + 0] * inv;
    x[2] = sums[(long)g * 4 + 1] * inv;
    x[3] = sums[(long)g * 4 + 2] * inv;
    x[4] = sums[(long)g * 4 + 3] * inv;
    x[5] = 0.0f; x[6] = 0.0f; x[7] = 0.0f;
}

// ---------------------------------------------------------------------------
// 3) GEMM1 via V_WMMA_F32_16X16X4_F32: H1 = X @ W1^T + b1   (K=5 padded to 8)
//    Also accumulates per-channel sum / sumsq for BN1 (LDS reduce -> global).
//    block = 256 (8 waves), one 16-token M-tile per wave, 4 N-tiles of 16.
// ---------------------------------------------------------------------------
__global__ void k_gemm1(const float* __restrict__ X,
                        const float* __restrict__ W,    // [64,5]
                        const float* __restrict__ bias, // [64]
                        float* __restrict__ H1,         // [Ntok,64]
                        float* __restrict__ stat) {     // [128] sum|sumsq
    __shared__ float s_sum[64], s_sq[64];
    if (threadIdx.x < 64) { s_sum[threadIdx.x] = 0.0f; s_sq[threadIdx.x] = 0.0f; }
    __syncthreads();

    const int wave  = threadIdx.x >> 5;
    const int lane  = threadIdx.x & 31;
    const int tileM = blockIdx.x * 8 + wave;
    const int row   = tileM * 16 + (lane & 15);
    const int khalf = (lane < 16) ? 0 : 2;         // A lanes 0-15: K 0,1 ; 16-31: K 2,3

    const v2f a0 = *(const v2f*)(X + (long)row * 8 + khalf);       // K 0..3
    const v2f a1 = *(const v2f*)(X + (long)row * 8 + 4 + khalf);   // K 4..7 (padded)
    const int nl = lane & 15;
    const int mbase = tileM * 16 + ((lane < 16) ? 0 : 8);

    for (int nt = 0; nt < 4; ++nt) {
        const int n = nt * 16 + nl;
        v2f b0, b1;
        b0.x = W[n * 5 + khalf];                    // k = 0 or 2
        b0.y = W[n * 5 + khalf + 1];                // k = 1 or 3
        b1.x = (khalf == 0) ? W[n * 5 + 4] : 0.0f;  // k = 4 (k=6 padded)
        b1.y = 0.0f;                                // k = 5 or 7 padded
        v8f c = {};
        c = __builtin_amdgcn_wmma_f32_16x16x4_f32(false, a0, false, b0,
                                                  (short)0, c, false, false);
        c = __builtin_amdgcn_wmma_f32_16x16x4_f32(false, a1, false, b1,
                                                  (short)0, c, false, false);
        const float bn = bias[n];
        float lsum = 0.0f, lsq = 0.0f;
        #pragma unroll
        for (int r = 0; r < 8; ++r) {
            const float v = c[r] + bn;
            H1[(long)(mbase + r) * 64 + n] = v;
            lsum += v; lsq += v * v;
        }
        atomicAdd(&s_sum[n], lsum);
        atomicAdd(&s_sq[n],  lsq);
    }
    __syncthreads();
    if (threadIdx.x < 64) {
        atomicAdd(&stat[threadIdx.x],      s_sum[threadIdx.x]);
        atomicAdd(&stat[64 + threadIdx.x], s_sq[threadIdx.x]);
    }
}

// ---------------------------------------------------------------------------
// 4) fold BN stats into affine: a = g*rsqrt(var+eps), c = b - mean*a
// ---------------------------------------------------------------------------
__global__ void k_coef(const float* __restrict__ stat,
                       const float* __restrict__ g,
                       const float* __restrict__ b,
                       float* __restrict__ coef, int C, float invN) {
    const int c = threadIdx.x;
    if (c < C) {
        const float mean = stat[c] * invN;
        const float var  = stat[C + c] * invN - mean * mean;
        const float a    = g[c] * rsqrtf(var + 1e-5f);
        coef[c]     = a;
        coef[C + c] = b[c] - mean * a;
    }
}

// ---------------------------------------------------------------------------
// 5) GEMM2 via WMMA: H2 = relu(bn1(H1)) @ W2^T + b2   (K=64 -> 16 wmma steps)
//    BN1+ReLU fused into the A-fragment build; BN2 stats accumulated.
// ---------------------------------------------------------------------------
__global__ void k_gemm2(const float* __restrict__ H1,
                        const float* __restrict__ coef1, // [128] a|c
                        const float* __restrict__ W,     // [32,64]
                        const float* __restrict__ bias,  // [32]
                        float* __restrict__ H2,          // [Ntok,32]
                        float* __restrict__ stat) {      // [64] sum|sumsq
    __shared__ float s_sum[32], s_sq[32];
    __shared__ float s_a[64], s_c[64];
    if (threadIdx.x < 32) { s_sum[threadIdx.x] = 0.0f; s_sq[threadIdx.x] = 0.0f; }
    if (threadIdx.x < 64) {
        s_a[threadIdx.x] = coef1[threadIdx.x];
        s_c[threadIdx.x] = coef1[64 + threadIdx.x];
    }
    __syncthreads();

    const int wave  = threadIdx.x >> 5;
    const int lane  = threadIdx.x & 31;
    const int tileM = blockIdx.x * 8 + wave;
    const int row   = tileM * 16 + (lane & 15);
    const int khalf = (lane < 16) ? 0 : 2;

    v2f afr[16];
    #pragma unroll
    for (int kb = 0; kb < 16; ++kb) {
        const int k0 = kb * 4 + khalf;
        const v2f h = *(const v2f*)(H1 + (long)row * 64 + k0);
        afr[kb].x = fmaxf(s_a[k0]     * h.x + s_c[k0],     0.0f);
        afr[kb].y = fmaxf(s_a[k0 + 1] * h.y + s_c[k0 + 1], 0.0f);
    }

    const int nl = lane & 15;
    const int mbase = tileM * 16 + ((lane < 16) ? 0 : 8);
    for (int nt = 0; nt < 2; ++nt) {
        const int n = nt * 16 + nl;
        v8f c = {};
        #pragma unroll
        for (int kb = 0; kb < 16; ++kb) {
            const int k0 = kb * 4 + khalf;
            const v2f bfr = *(const v2f*)(W + (long)n * 64 + k0);
            c = __builtin_amdgcn_wmma_f32_16x16x4_f32(false, afr[kb], false, bfr,
                                                      (short)0, c, false, false);
        }
        const float bn = bias[n];
        float lsum = 0.0f, lsq = 0.0f;
        #pragma unroll
        for (int r = 0; r < 8; ++r) {
            const float v = c[r] + bn;
            H2[(long)(mbase + r) * 32 + n] = v;
            lsum += v; lsq += v * v;
        }
        atomicAdd(&s_sum[n], lsum);
        atomicAdd(&s_sq[n],  lsq);
    }
    __syncthreads();
    if (threadIdx.x < 32) {
        atomicAdd(&stat[threadIdx.x],      s_sum[threadIdx.x]);
        atomicAdd(&stat[32 + threadIdx.x], s_sq[threadIdx.x]);
    }
}

// ---------------------------------------------------------------------------
// 6) BN2 + ReLU + row L2-normalize -> out[Ntok,32]
// ---------------------------------------------------------------------------
__global__ void k_final(const float* __restrict__ H2,
                        const float* __restrict__ coef2, // [64] a|c
                        float* __restrict__ out, int Ntok) {
    const int t = blockIdx.x * blockDim.x + threadIdx.x;
    if (t >= Ntok) return;
    float y[32];
    float ss = 0.0f;
    #pragma unroll
    for (int c = 0; c < 32; ++c) {
        const float v = fmaxf(coef2[c] * H2[(long)t * 32 + c] + coef2[32 + c], 0.0f);
        y[c] = v; ss += v * v;
    }
    const float inv = 1.0f / fmaxf(sqrtf(ss), 1e-8f);
    #pragma unroll
    for (int c = 0; c < 32; ++c) out[(long)t * 32 + c] = y[c] * inv;
}

// ---------------------------------------------------------------------------
// host orchestration
// ---------------------------------------------------------------------------
extern "C" void kernel_launch(void* const* d_in, const int* in_sizes, int n_in,
                              void* d_out, int out_size, void* d_ws, size_t ws_size,
                              hipStream_t stream) {
    // input order: labels, fx, fy, frame_idx, n_frames, n_labels,
    //              conv_w, conv_b, bn1_g, bn1_b, lin_w, lin_b, bn2_g, bn2_b
    const int*   labels    = (const int*)  d_in[0];
    const float* fx        = (const float*)d_in[1];
    const float* fy        = (const float*)d_in[2];
    const int*   frame_idx = (const int*)  d_in[3];
    const int*   n_frames  = (const int*)  d_in[4];
    const float* conv_w    = (const float*)d_in[6];
    const float* conv_b    = (const float*)d_in[7];
    const float* bn1_g     = (const float*)d_in[8];
    const float* bn1_b     = (const float*)d_in[9];
    const float* lin_w     = (const float*)d_in[10];
    const float* lin_b     = (const float*)d_in[11];
    const float* bn2_g     = (const float*)d_in[12];
    const float* bn2_b     = (const float*)d_in[13];
    float* out = (float*)d_out;

    const int B    = in_sizes[3];          // 16
    const int Ntok = out_size / 32;        // B*K = 16384
    const int K    = Ntok / B;             // 1024
    const int P    = in_sizes[0] / B;      // 262144
    int side = 1;                          // sqrt(P): labels are (B,1,side,side)
    while ((long)(side + 1) * (side + 1) <= (long)P) ++side;

    // workspace layout (floats)
    float* w = (float*)d_ws;
    long o = 0;
    float* sums   = w + o; o += (long)Ntok * 4;
    float* counts = w + o; o += Ntok;
    float* stat1  = w + o; o += 128;
    float* stat2  = w + o; o += 64;
    const long zeroN = o;                  // everything above must start at 0
    float* X      = w + o; o += (long)Ntok * 8;
    float* H1     = w + o; o += (long)Ntok * 64;
    float* H2     = w + o; o += (long)Ntok * 32;
    float* coef1  = w + o; o += 128;
    float* coef2  = w + o; o += 64;
    (void)ws_size; (void)n_in;

    // 0) zero accumulators
    k_zero<<<dim3((unsigned)((zeroN + 255) / 256)), dim3(256), 0, stream>>>(w, zeroN);

    // 1) pooling: 16 chunks per image, 20KB LDS accumulator per block
    const int chunks = 16;
    k_pool<<<dim3(chunks, B), dim3(256), K * 5 * sizeof(float), stream>>>(
        labels, fx, fy, sums, counts, K, P, side, chunks);

    // 2) build padded feature matrix
    k_buildX<<<dim3((Ntok + 255) / 256), dim3(256), 0, stream>>>(
        sums, counts, frame_idx, n_frames, X, K, Ntok);

    // 3) GEMM1 (WMMA f32) + BN1 stats
    k_gemm1<<<dim3(Ntok / 128), dim3(256), 0, stream>>>(X, conv_w, conv_b, H1, stat1);

    // 4) BN1 affine fold
    k_coef<<<dim3(1), dim3(64), 0, stream>>>(stat1, bn1_g, bn1_b, coef1, 64,
                                             1.0f / (float)Ntok);

    // 5) GEMM2 (WMMA f32, BN1+ReLU fused on A) + BN2 stats
    k_gemm2<<<dim3(Ntok / 128), dim3(256), 0, stream>>>(H1, coef1, lin_w, lin_b, H2, stat2);

    // 6) BN2 affine fold
    k_coef<<<dim3(1), dim3(32), 0, stream>>>(stat2, bn2_g, bn2_b, coef2, 32,
                                             1.0f / (float)Ntok);

    // 7) BN2 + ReLU + L2 normalize
    k_final<<<dim3((Ntok + 255) / 256), dim3(256), 0, stream>>>(H2, coef2, out, Ntok);
}